// DBCHCN_62019327754646
// MI455X (gfx1250) — compile-verified
//
#include <hip/hip_runtime.h>
#include <math.h>

typedef __attribute__((ext_vector_type(2))) float v2f;
typedef __attribute__((ext_vector_type(8))) float v8f;

#define C_IN      64
#define T_DIM     128
#define V_DIM     25
#define TT        16                 // t-tile size
#define P_STR     28                 // padded v-row stride (25 -> 28, K-pad for mix WMMA)
#define CH_LDS    (TT * P_STR)       // 448 floats per channel in LDS stage
#define CH_STRIDE (T_DIM * V_DIM)    // 3200
#define M_STR     28                 // padded mix-matrix row stride (32 rows x 28 cols)
#define M_ELEMS   (32 * M_STR)       // 896

// ---------------------------------------------------------------------------
// Phase A: x8t[n,8,T] = W7 * mean_v(x) + b7 ; x9v[n,8,V] = W8 * mean_t(x) + b8
// (mean of conv == conv of mean, since conv1x1 is linear over positions)
// ---------------------------------------------------------------------------
__global__ __launch_bounds__(256) void phaseA_kernel(
    const float* __restrict__ x,
    const float* __restrict__ W7, const float* __restrict__ b7,
    const float* __restrict__ W8, const float* __restrict__ b8,
    float* __restrict__ x8t, float* __restrict__ x9v)
{
    __shared__ float xv[C_IN * T_DIM];  // mean over v  [i][t]
    __shared__ float xt[C_IN * V_DIM];  // mean over t  [i][v]
    const int n = blockIdx.x;
    const int tid = threadIdx.x;
    const float* xn = x + (size_t)n * C_IN * CH_STRIDE;

    for (int idx = tid; idx < C_IN * T_DIM; idx += 256) {
        const float* p = xn + (size_t)idx * V_DIM;   // idx = i*128 + t
        float s = 0.f;
        #pragma unroll
        for (int v = 0; v < V_DIM; ++v) s += p[v];
        xv[idx] = s * (1.0f / V_DIM);
    }
    for (int idx = tid; idx < C_IN * V_DIM; idx += 256) {
        const int i = idx / V_DIM, v = idx % V_DIM;
        const float* p = xn + (size_t)i * CH_STRIDE + v;
        float s = 0.f;
        for (int t = 0; t < T_DIM; ++t) s += p[t * V_DIM];
        xt[idx] = s * (1.0f / T_DIM);
    }
    __syncthreads();
    for (int idx = tid; idx < 8 * T_DIM; idx += 256) {
        const int r = idx >> 7, t = idx & 127;
        float s = b7[r];
        #pragma unroll 8
        for (int i = 0; i < C_IN; ++i) s += W7[r * C_IN + i] * xv[i * T_DIM + t];
        x8t[(size_t)(n * 8 + r) * T_DIM + t] = s;
    }
    for (int idx = tid; idx < 8 * V_DIM; idx += 256) {
        const int r = idx / V_DIM, v = idx % V_DIM;
        float s = b8[r];
        #pragma unroll 8
        for (int i = 0; i < C_IN; ++i) s += W8[r * C_IN + i] * xt[i * V_DIM + v];
        x9v[(size_t)n * 8 * V_DIM + idx] = s;
    }
}

// ---------------------------------------------------------------------------
// 64x64 channel GEMM over one (n, t-tile): dst[c][tl][v] = sum_i W[c,i]*x[i,p] + b[c]
// WMMA f32 16x16x4: A = W tile (M=outch,K=inch), B = x tile (K=inch,N=pos).
// A layout: lane L -> row M=L%16; VGPR j, half h=L/16 -> K = 2h+j.
// B layout (mirrors A K-order): VGPR j, half h -> row K = 2h+j, col N = p0+L%16.
// C/D: VGPR r, half h -> M = r + 8h, N = L%16.
// dst is the padded LDS stage (stride P_STR per t-row; pads are never written).
// ---------------------------------------------------------------------------
__device__ __forceinline__ void gemm64_wmma(
    const float* __restrict__ W, const float* __restrict__ b,
    const float* __restrict__ xn, int t0, float* __restrict__ dst,
    int wave, int lane)
{
    const int l16 = lane & 15, h = lane >> 4;
    for (int tile = wave; tile < 100; tile += 8) {           // 4 M-tiles x 25 N-tiles
        const int m0 = (tile / 25) * 16;
        const int p0 = (tile % 25) * 16;
        const int p  = p0 + l16;
        const int tl = p / V_DIM, v = p % V_DIM;
        const float* xcol = xn + (size_t)(t0 + tl) * V_DIM + v;  // + i*CH_STRIDE
        v8f acc = {0.f, 0.f, 0.f, 0.f, 0.f, 0.f, 0.f, 0.f};
        #pragma unroll
        for (int kc = 0; kc < 16; ++kc) {
            const int k0 = kc * 4 + 2 * h;
            v2f a, bf;
            a.x = W[(m0 + l16) * C_IN + k0];
            a.y = W[(m0 + l16) * C_IN + k0 + 1];
            bf.x = xcol[(size_t)k0 * CH_STRIDE];
            bf.y = xcol[(size_t)(k0 + 1) * CH_STRIDE];
            acc = __builtin_amdgcn_wmma_f32_16x16x4_f32(false, a, false, bf,
                                                        (short)0, acc, false, false);
        }
        #pragma unroll
        for (int r = 0; r < 8; ++r) {
            const int c = m0 + r + 8 * h;
            dst[c * CH_LDS + tl * P_STR + v] = acc[r] + b[c];
        }
    }
}

// z9 = W9[64,8] * rel + b9, rel(k,p) = tanh(x8t[k,t] - x9v[k,v]) generated on the fly.
__device__ __forceinline__ void gemmZ9_wmma(
    const float* __restrict__ W9, const float* __restrict__ b9,
    const float* __restrict__ x8ts, const float* __restrict__ x9vs,
    float* __restrict__ dst, int wave, int lane)
{
    const int l16 = lane & 15, h = lane >> 4;
    for (int tile = wave; tile < 100; tile += 8) {
        const int m0 = (tile / 25) * 16;
        const int p0 = (tile % 25) * 16;
        const int p  = p0 + l16;
        const int tl = p / V_DIM, v = p % V_DIM;
        v8f acc = {0.f, 0.f, 0.f, 0.f, 0.f, 0.f, 0.f, 0.f};
        #pragma unroll
        for (int kc = 0; kc < 2; ++kc) {                      // K = 8
            const int k0 = kc * 4 + 2 * h;
            v2f a, bf;
            a.x = W9[(m0 + l16) * 8 + k0];
            a.y = W9[(m0 + l16) * 8 + k0 + 1];
            bf.x = tanhf(x8ts[k0 * TT + tl]       - x9vs[k0 * V_DIM + v]);
            bf.y = tanhf(x8ts[(k0 + 1) * TT + tl] - x9vs[(k0 + 1) * V_DIM + v]);
            acc = __builtin_amdgcn_wmma_f32_16x16x4_f32(false, a, false, bf,
                                                        (short)0, acc, false, false);
        }
        #pragma unroll
        for (int r = 0; r < 8; ++r) {
            const int c = m0 + r + 8 * h;
            dst[c * CH_LDS + tl * P_STR + v] = acc[r] + b9[c];
        }
    }
}

// ---------------------------------------------------------------------------
// 25x25 V-axis mix on the matrix pipe: D[v,(c,tl)] = sum_u Mp[v,u] * ys[c][tl][u]
// Mp is zero-padded to 32x28 (rows M, cols K); ys rows are zero-padded to 28.
// N-tiles map 1:1 to channels: nt == c, lane's l16 == tl. 2 M-tiles x 7 K-steps.
// A-fragments (7x2 regs) hoisted and reused across all 64 N-tiles.
// Epilogue lambda gets acc, m0, c(=nt), h, l16(=tl); rows v = m0+r+8h (<25 valid).
// ---------------------------------------------------------------------------
template <typename EPI>
__device__ __forceinline__ void mix_wmma(
    const float* __restrict__ Mp,   // [32][M_STR] zero-padded mix matrix in LDS
    const float* __restrict__ ys,   // [64][CH_LDS] padded stage in LDS
    int wave, int lane, EPI epi)
{
    const int l16 = lane & 15, h = lane >> 4;
    #pragma unroll
    for (int mh = 0; mh < 2; ++mh) {
        const int m0 = mh * 16;
        float Ax[7], Ay[7];
        #pragma unroll
        for (int kc = 0; kc < 7; ++kc) {
            const int u0 = kc * 4 + 2 * h;
            Ax[kc] = Mp[(m0 + l16) * M_STR + u0];
            Ay[kc] = Mp[(m0 + l16) * M_STR + u0 + 1];
        }
        for (int nt = wave; nt < C_IN; nt += 8) {
            const float* yrow = ys + nt * CH_LDS + l16 * P_STR;
            v8f acc = {0.f, 0.f, 0.f, 0.f, 0.f, 0.f, 0.f, 0.f};
            #pragma unroll
            for (int kc = 0; kc < 7; ++kc) {
                const int u0 = kc * 4 + 2 * h;
                v2f a, bf;
                a.x = Ax[kc]; a.y = Ay[kc];
                bf.x = yrow[u0]; bf.y = yrow[u0 + 1];
                acc = __builtin_amdgcn_wmma_f32_16x16x4_f32(false, a, false, bf,
                                                            (short)0, acc, false, false);
            }
            epi(acc, m0, nt, h, l16);
        }
    }
}

// ---------------------------------------------------------------------------
// Phase B: one block per (n, 16-wide t tile). Dynamic LDS:
//   ys[64*448] | mb[64*448] | Ap[896] | A5p[896] | vtp[896] | x8ts[128] | x9vs[200]
// ---------------------------------------------------------------------------
__global__ __launch_bounds__(256) void phaseB_kernel(
    const float* __restrict__ x,
    const float* __restrict__ A,  const float* __restrict__ A5,
    const float* __restrict__ W1, const float* __restrict__ b1,
    const float* __restrict__ W3, const float* __restrict__ b3,
    const float* __restrict__ W5, const float* __restrict__ b5,
    const float* __restrict__ W6, const float* __restrict__ b6,
    const float* __restrict__ W9, const float* __restrict__ b9,
    const float* __restrict__ x8t, const float* __restrict__ x9v,
    float* __restrict__ out)
{
    extern __shared__ float sm[];
    float* ys   = sm;                        // 64*448
    float* mb   = ys + C_IN * CH_LDS;        // 64*448
    float* Ap   = mb + C_IN * CH_LDS;        // 896
    float* A5p  = Ap + M_ELEMS;              // 896
    float* vtp  = A5p + M_ELEMS;             // 896
    float* x8ts = vtp + M_ELEMS;             // 128
    float* x9vs = x8ts + 8 * TT;             // 200

    const int n  = blockIdx.x >> 3;
    const int t0 = (blockIdx.x & 7) * TT;
    const int tid = threadIdx.x;
    const int wave = tid >> 5, lane = tid & 31;
    const float* xn = x + (size_t)n * C_IN * CH_STRIDE;
    const size_t obase = (size_t)n * 256 * CH_STRIDE;

    // S0: zero padded regions once (GEMM epilogues never write pads), stage matrices
    for (int i = tid; i < C_IN * CH_LDS; i += 256) ys[i] = 0.f;
    for (int i = tid; i < 3 * M_ELEMS; i += 256) Ap[i] = 0.f;
    for (int i = tid; i < 8 * TT; i += 256)
        x8ts[i] = x8t[(size_t)(n * 8 + (i >> 4)) * T_DIM + t0 + (i & 15)];
    for (int i = tid; i < 8 * V_DIM; i += 256)
        x9vs[i] = x9v[(size_t)n * 8 * V_DIM + i];
    __syncthreads();
    for (int i = tid; i < V_DIM * V_DIM; i += 256) {
        const int r = i / V_DIM, cidx = i % V_DIM;
        Ap[r * M_STR + cidx]  = A[i];
        A5p[r * M_STR + cidx] = A5[i];
    }
    __syncthreads();
    for (int i = tid; i < V_DIM * V_DIM; i += 256) {  // vt[u,v]=tanh(meancol(A)[v]-meanrow(A5)[u])
        const int u = i / V_DIM, v = i % V_DIM;
        float am = 0.f, a5m = 0.f;
        #pragma unroll
        for (int r = 0; r < V_DIM; ++r) { am += Ap[r * M_STR + v]; a5m += A5p[u * M_STR + r]; }
        vtp[u * M_STR + v] = tanhf((am - a5m) * (1.0f / V_DIM));
    }
    __syncthreads();

    // S1: y6 -> ys ; mix7a = A5*y6 -> mb (WMMA mix)
    gemm64_wmma(W6, b6, xn, t0, ys, wave, lane);
    __syncthreads();
    mix_wmma(A5p, ys, wave, lane,
        [&](const v8f& acc, int m0, int c, int h, int tl) {
            #pragma unroll
            for (int r = 0; r < 8; ++r) {
                const int v = m0 + r + 8 * h;
                if (v < V_DIM) mb[c * CH_LDS + tl * P_STR + v] = acc[r];
            }
        });
    __syncthreads();

    // S2: y1 -> ys ; x1 = A*y1 ; x7 = (x1 + mix7a)/2 ; keep x7 in mb
    gemm64_wmma(W1, b1, xn, t0, ys, wave, lane);
    __syncthreads();
    mix_wmma(Ap, ys, wave, lane,
        [&](const v8f& acc, int m0, int c, int h, int tl) {
            const size_t oo = obase + (size_t)c * CH_STRIDE + (size_t)(t0 + tl) * V_DIM;
            #pragma unroll
            for (int r = 0; r < 8; ++r) {
                const int v = m0 + r + 8 * h;
                if (v < V_DIM) {
                    const float x1v = acc[r];
                    const float x7v = 0.5f * (x1v + mb[c * CH_LDS + tl * P_STR + v]);
                    out[oo + v] = x1v;                              // channels [0,64)
                    out[oo + (size_t)64 * CH_STRIDE + v] = x7v;     // channels [64,128)
                    mb[c * CH_LDS + tl * P_STR + v] = x7v;
                }
            }
        });
    __syncthreads();

    // S3: y3 -> ys ; x3 = (vt*y3 + x7)/2
    gemm64_wmma(W3, b3, xn, t0, ys, wave, lane);
    __syncthreads();
    mix_wmma(vtp, ys, wave, lane,
        [&](const v8f& acc, int m0, int c, int h, int tl) {
            const size_t oo = obase + (size_t)(192 + c) * CH_STRIDE + (size_t)(t0 + tl) * V_DIM;
            #pragma unroll
            for (int r = 0; r < 8; ++r) {
                const int u = m0 + r + 8 * h;
                if (u < V_DIM)
                    out[oo + u] = 0.5f * (acc[r] + mb[c * CH_LDS + tl * P_STR + u]);
            }
        });
    __syncthreads();

    // S4/S5: y5 -> mb ; z9 = W9*rel -> ys
    gemm64_wmma(W5, b5, xn, t0, mb, wave, lane);
    gemmZ9_wmma(W9, b9, x8ts, x9vs, ys, wave, lane);
    __syncthreads();

    // S6: x8 = (A5*z9) * y5
    mix_wmma(A5p, ys, wave, lane,
        [&](const v8f& acc, int m0, int c, int h, int tl) {
            const size_t oo = obase + (size_t)(128 + c) * CH_STRIDE + (size_t)(t0 + tl) * V_DIM;
            #pragma unroll
            for (int r = 0; r < 8; ++r) {
                const int v = m0 + r + 8 * h;
                if (v < V_DIM)
                    out[oo + v] = acc[r] * mb[c * CH_LDS + tl * P_STR + v];
            }
        });
}

extern "C" void kernel_launch(void* const* d_in, const int* in_sizes, int n_in,
                              void* d_out, int out_size, void* d_ws, size_t ws_size,
                              hipStream_t stream) {
    (void)in_sizes; (void)n_in; (void)out_size; (void)ws_size;
    const float* x  = (const float*)d_in[0];
    const float* A  = (const float*)d_in[1];
    const float* A5 = (const float*)d_in[2];
    const float* W1 = (const float*)d_in[3];  const float* b1 = (const float*)d_in[4];
    const float* W3 = (const float*)d_in[5];  const float* b3 = (const float*)d_in[6];
    const float* W5 = (const float*)d_in[7];  const float* b5 = (const float*)d_in[8];
    const float* W6 = (const float*)d_in[9];  const float* b6 = (const float*)d_in[10];
    const float* W7 = (const float*)d_in[11]; const float* b7 = (const float*)d_in[12];
    const float* W8 = (const float*)d_in[13]; const float* b8 = (const float*)d_in[14];
    const float* W9 = (const float*)d_in[15]; const float* b9 = (const float*)d_in[16];
    float* out = (float*)d_out;

    float* ws_x8t = (float*)d_ws;                 // 64*8*128 floats
    float* ws_x9v = ws_x8t + 64 * 8 * 128;        // 64*8*25 floats

    phaseA_kernel<<<64, 256, 0, stream>>>(x, W7, b7, W8, b8, ws_x8t, ws_x9v);

    const size_t smem = (size_t)(2 * C_IN * CH_LDS + 3 * M_ELEMS + 8 * TT + 8 * V_DIM)
                        * sizeof(float);
    phaseB_kernel<<<64 * 8, 256, smem, stream>>>(x, A, A5, W1, b1, W3, b3, W5, b5,
                                                 W6, b6, W9, b9, ws_x8t, ws_x9v, out);
}